// Decoder_11613591569166
// MI455X (gfx1250) — compile-verified
//
#include <hip/hip_runtime.h>
#include <math.h>

typedef __attribute__((ext_vector_type(2))) float v2f;
typedef __attribute__((ext_vector_type(4))) float v4f;
typedef __attribute__((ext_vector_type(8))) float v8f;

// ---------------- problem constants (from reference) ----------------
#define NB        131072      // batch
// workspace layout (floats)
#define OFF_WLAT  0           // 16 nt * 64 kt * 64 = 65536
#define OFF_W0    65536       //  8 nt * 64 kt * 64 = 32768
#define OFF_W1    98304       //  7 nt * 52 kt * 64 = 23296   (O 104 -> pad 112)
#define OFF_W2    121600      //  6 nt * 46 kt * 64 = 17664   (O  92 -> pad  96)
#define OFF_BLAT  139264      // 256
#define OFF_B0    139520      // 128
#define OFF_B1    139648      // 112 (padded)
#define OFF_B2    139760      //  96 (padded)
#define WS_FLOATS 139856
// LDS staging pitches (words); 260 % 64 == 4, 132 % 64 == 4 -> conflict-free
#define PITCH_A   260
#define PITCH_B   132
#define MT        32          // batch rows per wave (2 WMMA M-tiles)

// ---- skeleton neighbor masks (precomputed: Floyd-Warshall dist<=2 + disp node) ----
__constant__ unsigned MASK0c[8]  = {0xFFu,0x8Fu,0x8Fu,0xFFu,0xF9u,0xF9u,0xF9u,0xFFu};
__constant__ unsigned MASK1c[13] = {0x107Fu,0x101Fu,0x102Fu,0x13CFu,0x1013u,0x1025u,0x1FC9u,
                                    0x03C8u,0x07C8u,0x0BC8u,0x0540u,0x0A40u,0x107Fu};
__constant__ unsigned MASK2c[23] = {0x40007Fu,0x40009Fu,0x40012Fu,0x40024Fu,0x400493u,0x400925u,
                                    0x407249u,0x000492u,0x000924u,0x03F248u,0x000490u,0x000920u,
                                    0x00F240u,0x057240u,0x0A7240u,0x009200u,0x152200u,0x2A4200u,
                                    0x152000u,0x2A4000u,0x150000u,0x2A0000u,0x40007Fu};
// unpool node maps packed as 4-bit nibbles
// UMAP1[13] = {0,1,2,3,1,2,3,4,5,6,5,6,7}
#define UM1    0x0007656543213210ULL
// UMAP2[23] = {0,1,2,3,1,2,3,4,5,6,4,5,7,8,9,7,8,9,10,11,10,11,12}
#define UM2LO  0x7987546543213210ULL
#define UM2HI  0x0000000000CBABA98ULL

// ---------------------------------------------------------------------------
// Prep: fold center conv tap (k=1) * skeleton mask into WMMA B-tile layout.
// B-tile (nt,kt) = 64 floats: lane L comp c -> W[nt*16+(L&15)][kt*4+2*(L>>4)+c]
// ---------------------------------------------------------------------------
__global__ __launch_bounds__(256) void prep_kernel(
    const float* __restrict__ wlat, const float* __restrict__ blat,
    const float* __restrict__ w0,   const float* __restrict__ b0,
    const float* __restrict__ w1,   const float* __restrict__ b1,
    const float* __restrict__ w2,   const float* __restrict__ b2,
    float* __restrict__ ws)
{
    int id = blockIdx.x * 256 + threadIdx.x;
    if (id >= WS_FLOATS) return;

    if (id < OFF_W0) {                       // w_lat, no mask, dense (256x256)
        int rel = id - OFF_WLAT;
        int block = rel >> 6, li = rel & 63;
        int lane = li >> 1, c = li & 1;
        int nt = block >> 6, kt = block & 63;
        int n = nt * 16 + (lane & 15);
        int k = kt * 4 + ((lane >> 4) << 1) + c;
        ws[id] = wlat[n * 256 + k];
        return;
    }
    if (id < OFF_W1) {                       // w0: (128,256,3) center tap, mask0
        int rel = id - OFF_W0;
        int block = rel >> 6, li = rel & 63;
        int lane = li >> 1, c = li & 1;
        int nt = block >> 6, kt = block & 63;
        int n = nt * 16 + (lane & 15);
        int k = kt * 4 + ((lane >> 4) << 1) + c;
        int mo = n >> 4, mi = k >> 5;        // cout=16, cin=32
        float v = ((MASK0c[mo] >> mi) & 1u) ? w0[(n * 256 + k) * 3 + 1] : 0.f;
        ws[id] = v;
        return;
    }
    if (id < OFF_W2) {                       // w1: (104,208,3), pad O->112, mask1
        int rel = id - OFF_W1;
        int block = rel >> 6, li = rel & 63;
        int lane = li >> 1, c = li & 1;
        int nt = block / 52, kt = block - nt * 52;
        int n = nt * 16 + (lane & 15);
        int k = kt * 4 + ((lane >> 4) << 1) + c;
        float v = 0.f;
        if (n < 104) {
            int mo = n >> 3, mi = k >> 4;    // cout=8, cin=16
            if ((MASK1c[mo] >> mi) & 1u) v = w1[(n * 208 + k) * 3 + 1];
        }
        ws[id] = v;
        return;
    }
    if (id < OFF_BLAT) {                     // w2: (92,184,3), pad O->96, mask2
        int rel = id - OFF_W2;
        int block = rel >> 6, li = rel & 63;
        int lane = li >> 1, c = li & 1;
        int nt = block / 46, kt = block - nt * 46;
        int n = nt * 16 + (lane & 15);
        int k = kt * 4 + ((lane >> 4) << 1) + c;
        float v = 0.f;
        if (n < 92) {
            int mo = n >> 2, mi = k >> 3;    // cout=4, cin=8
            if ((MASK2c[mo] >> mi) & 1u) v = w2[(n * 184 + k) * 3 + 1];
        }
        ws[id] = v;
        return;
    }
    if (id < OFF_B0) { ws[id] = blat[id - OFF_BLAT]; return; }
    if (id < OFF_B1) { ws[id] = b0[id - OFF_B0]; return; }
    if (id < OFF_B2) { int i = id - OFF_B1; ws[id] = (i < 104) ? b1[i] : 0.f; return; }
    { int i = id - OFF_B2; ws[id] = (i < 92) ? b2[i] : 0.f; }
}

// ---------------------------------------------------------------------------
// Fused decoder: one wave = 32 batch rows (2 M-tiles sharing every B-operand
// load), full 4-GEMM chain + quaternion epilogue.
// A operands: LDS (ds_load_b64) / global for the first layer.
// B operands: pre-packed coalesced global (L2/WGP$-resident) -> split-source
// keeps LDS and VMEM ports both below saturation.
// ---------------------------------------------------------------------------
__global__ __launch_bounds__(32) void decoder_fused(
    const float* __restrict__ x_in,
    const float* __restrict__ mean_dqs,
    const float* __restrict__ std_dqs,
    const float* __restrict__ ws,
    float* __restrict__ out)
{
    __shared__ float buf0[MT * PITCH_A];   // 33.3 KB
    __shared__ float buf1[MT * PITCH_B];   // 16.9 KB
    const int lane = threadIdx.x & 31;
    const int row  = lane & 15;            // M-row for A; N-col for B/C/D
    const int kh   = lane >> 4;            // K-half selector
    const int rbase = blockIdx.x * MT;

    // ---------------- layer "lat": Y = X @ Wlat^T + b  -> buf0 (256 cols)
    {
        const float* xr0 = x_in + (size_t)(rbase + row) * 256 + 2 * kh;
        const float* xr1 = xr0 + 16 * 256;
        const v2f* wB = (const v2f*)(ws + OFF_WLAT) + lane;
        for (int nt = 0; nt < 16; ++nt) {
            float bv = ws[OFF_BLAT + nt * 16 + row];
            v8f c0 = {bv, bv, bv, bv, bv, bv, bv, bv};
            v8f c1 = c0;
            const v2f* wp = wB + nt * 64 * 32;
#pragma unroll 4
            for (int kt = 0; kt < 64; ++kt) {
                v2f b  = wp[kt * 32];
                v2f a0 = *(const v2f*)(xr0 + kt * 4);
                v2f a1 = *(const v2f*)(xr1 + kt * 4);
                c0 = __builtin_amdgcn_wmma_f32_16x16x4_f32(false, a0, false, b,
                                                           (short)0, c0, false, false);
                c1 = __builtin_amdgcn_wmma_f32_16x16x4_f32(false, a1, false, b,
                                                           (short)0, c1, false, false);
            }
#pragma unroll
            for (int r = 0; r < 8; ++r) {
                buf0[(r + 8 * kh) * PITCH_A + nt * 16 + row]      = c0[r];
                buf0[(16 + r + 8 * kh) * PITCH_A + nt * 16 + row] = c1[r];
            }
        }
    }
    // ---------------- layer 0: 256 -> 128, identity unpool, LeakyReLU -> buf1
    {
        const v2f* wB = (const v2f*)(ws + OFF_W0) + lane;
        for (int nt = 0; nt < 8; ++nt) {
            float bv = ws[OFF_B0 + nt * 16 + row];
            v8f c0 = {bv, bv, bv, bv, bv, bv, bv, bv};
            v8f c1 = c0;
            const v2f* wp = wB + nt * 64 * 32;
#pragma unroll 4
            for (int kt = 0; kt < 64; ++kt) {
                int k0 = kt * 4 + 2 * kh;
                v2f b  = wp[kt * 32];
                v2f a0 = *(const v2f*)(buf0 + row * PITCH_A + k0);
                v2f a1 = *(const v2f*)(buf0 + (row + 16) * PITCH_A + k0);
                c0 = __builtin_amdgcn_wmma_f32_16x16x4_f32(false, a0, false, b,
                                                           (short)0, c0, false, false);
                c1 = __builtin_amdgcn_wmma_f32_16x16x4_f32(false, a1, false, b,
                                                           (short)0, c1, false, false);
            }
#pragma unroll
            for (int r = 0; r < 8; ++r) {
                float v0 = c0[r], v1 = c1[r];
                v0 = (v0 >= 0.f) ? v0 : 0.2f * v0;
                v1 = (v1 >= 0.f) ? v1 : 0.2f * v1;
                buf1[(r + 8 * kh) * PITCH_B + nt * 16 + row]      = v0;
                buf1[(16 + r + 8 * kh) * PITCH_B + nt * 16 + row] = v1;
            }
        }
    }
    // ---------------- layer 1: gather 128->208 (UMAP1, cin=16), GEMM 208->112, LeakyReLU -> buf0
    {
        const v2f* wB = (const v2f*)(ws + OFF_W1) + lane;
        for (int nt = 0; nt < 7; ++nt) {
            float bv = ws[OFF_B1 + nt * 16 + row];
            v8f c0 = {bv, bv, bv, bv, bv, bv, bv, bv};
            v8f c1 = c0;
            const v2f* wp = wB + nt * 52 * 32;
#pragma unroll 4
            for (int kt = 0; kt < 52; ++kt) {
                int k0 = kt * 4 + 2 * kh;
                int node = kt >> 2;          // K-half offset never crosses a node block
                int src = (int)((UM1 >> (node * 4)) & 0xF) * 16 + (k0 & 15);
                v2f b  = wp[kt * 32];
                v2f a0 = *(const v2f*)(buf1 + row * PITCH_B + src);
                v2f a1 = *(const v2f*)(buf1 + (row + 16) * PITCH_B + src);
                c0 = __builtin_amdgcn_wmma_f32_16x16x4_f32(false, a0, false, b,
                                                           (short)0, c0, false, false);
                c1 = __builtin_amdgcn_wmma_f32_16x16x4_f32(false, a1, false, b,
                                                           (short)0, c1, false, false);
            }
#pragma unroll
            for (int r = 0; r < 8; ++r) {
                float v0 = c0[r], v1 = c1[r];
                v0 = (v0 >= 0.f) ? v0 : 0.2f * v0;
                v1 = (v1 >= 0.f) ? v1 : 0.2f * v1;
                buf0[(r + 8 * kh) * PITCH_A + nt * 16 + row]      = v0;
                buf0[(16 + r + 8 * kh) * PITCH_A + nt * 16 + row] = v1;
            }
        }
    }
    // ---------------- layer 2: gather 104->184 (UMAP2, cin=8), GEMM 184->96 -> buf1
    {
        const v2f* wB = (const v2f*)(ws + OFF_W2) + lane;
        for (int nt = 0; nt < 6; ++nt) {
            float bv = ws[OFF_B2 + nt * 16 + row];
            v8f c0 = {bv, bv, bv, bv, bv, bv, bv, bv};
            v8f c1 = c0;
            const v2f* wp = wB + nt * 46 * 32;
#pragma unroll 4
            for (int kt = 0; kt < 46; ++kt) {
                int k0 = kt * 4 + 2 * kh;
                int node = kt >> 1;          // K-half offset never crosses a node block
                unsigned nib = (node < 16) ? (unsigned)((UM2LO >> (node * 4)) & 0xF)
                                           : (unsigned)((UM2HI >> ((node - 16) * 4)) & 0xF);
                int src = (int)nib * 8 + (k0 & 7);
                v2f b  = wp[kt * 32];
                v2f a0 = *(const v2f*)(buf0 + row * PITCH_A + src);
                v2f a1 = *(const v2f*)(buf0 + (row + 16) * PITCH_A + src);
                c0 = __builtin_amdgcn_wmma_f32_16x16x4_f32(false, a0, false, b,
                                                           (short)0, c0, false, false);
                c1 = __builtin_amdgcn_wmma_f32_16x16x4_f32(false, a1, false, b,
                                                           (short)0, c1, false, false);
            }
#pragma unroll
            for (int r = 0; r < 8; ++r) {
                buf1[(r + 8 * kh) * PITCH_B + nt * 16 + row]      = c0[r];
                buf1[(16 + r + 8 * kh) * PITCH_B + nt * 16 + row] = c1[r];
            }
        }
    }
    // ---------------- epilogue: quaternion renorm on cols 0..87, copy cols 88..90
    // motion: 32 rows * 22 quats = 704 items = 22 full wave iterations
#pragma unroll 1
    for (int it = 0; it < 22; ++it) {
        int idx = it * 32 + lane;
        int rw = idx / 22;
        int j  = idx - rw * 22;
        v4f v = *(const v4f*)(buf1 + rw * PITCH_B + 4 * j);
        v4f m = *(const v4f*)(mean_dqs + 8 * j);
        v4f s = *(const v4f*)(std_dqs + 8 * j);
        v4f t;
        t.x = v.x * s.x + m.x;
        t.y = v.y * s.y + m.y;
        t.z = v.z * s.z + m.z;
        t.w = v.w * s.w + m.w;
        float n2 = t.x * t.x + t.y * t.y + t.z * t.z + t.w * t.w;
        float inv = 1.0f / sqrtf(n2);
        v4f o;
        o.x = (t.x * inv - m.x) / s.x;
        o.y = (t.y * inv - m.y) / s.y;
        o.z = (t.z * inv - m.z) / s.z;
        o.w = (t.w * inv - m.w) / s.w;
        *(v4f*)(out + (size_t)(rbase + rw) * 88 + 4 * j) = o;
    }
    // displacement: 32 rows * 3 cols = 96 items = 3 full wave iterations
#pragma unroll 1
    for (int it = 0; it < 3; ++it) {
        int idx = it * 32 + lane;
        int rw = idx / 3, cc = idx - rw * 3;
        out[(size_t)NB * 88 + (size_t)(rbase + rw) * 3 + cc] =
            buf1[rw * PITCH_B + 88 + cc];
    }
}

// ---------------------------------------------------------------------------
extern "C" void kernel_launch(void* const* d_in, const int* in_sizes, int n_in,
                              void* d_out, int out_size, void* d_ws, size_t ws_size,
                              hipStream_t stream) {
    const float* x    = (const float*)d_in[0];
    const float* mean = (const float*)d_in[1];
    const float* stdv = (const float*)d_in[2];
    const float* wlat = (const float*)d_in[3];
    const float* blat = (const float*)d_in[4];
    const float* w0   = (const float*)d_in[5];
    const float* b0   = (const float*)d_in[6];
    const float* w1   = (const float*)d_in[7];
    const float* b1   = (const float*)d_in[8];
    const float* w2   = (const float*)d_in[9];
    const float* b2   = (const float*)d_in[10];
    float* ws  = (float*)d_ws;
    float* out = (float*)d_out;

    int prep_blocks = (WS_FLOATS + 255) / 256;
    prep_kernel<<<prep_blocks, 256, 0, stream>>>(wlat, blat, w0, b0, w1, b1, w2, b2, ws);
    decoder_fused<<<NB / MT, 32, 0, stream>>>(x, mean, stdv, ws, out);
}